// UnionDetector_52501680227030
// MI455X (gfx1250) — compile-verified
//
#include <hip/hip_runtime.h>
#include <stdint.h>

#define BATCH    16
#define MAX_LAB  64
#define NPB      (BATCH * MAX_LAB)     // 1024 people boxes
#define M_BOXES  50000
#define IMG_W    640.0f
#define IMG_H    640.0f
#define CHUNK    1024                  // boxes staged per LDS tile (16 KB)
#define TPB      256                   // 8 wave32

// int4 vector type matching the builtin's parameter type
typedef int v4i __attribute__((vector_size(16)));
typedef __attribute__((address_space(1))) v4i gv4i_t;   // global (prints __device__)
typedef __attribute__((address_space(3))) v4i lv4i_t;   // LDS    (prints __shared__)

// ---- CDNA5 async global->LDS copy (ASYNCcnt path) -------------------------
__device__ __forceinline__ void async_copy_b128(const void* g, void* l) {
#if __has_builtin(__builtin_amdgcn_global_load_async_to_lds_b128)
  __builtin_amdgcn_global_load_async_to_lds_b128((gv4i_t*)g, (lv4i_t*)l, 0, 0);
#else
  uint32_t lds_off = (uint32_t)(uintptr_t)(lv4i_t*)l;    // flat -> LDS offset
  asm volatile("global_load_async_to_lds_b128 %0, %1, off"
               :: "v"(lds_off), "v"((uint64_t)(uintptr_t)g)
               : "memory");
#endif
}

__device__ __forceinline__ void wait_async_zero() {
#if __has_builtin(__builtin_amdgcn_s_wait_asynccnt)
  __builtin_amdgcn_s_wait_asynccnt(0);
#else
  asm volatile("s_wait_asynccnt 0" ::: "memory");
#endif
}

// ---- kernel 0: zero the per-people-box running max ------------------------
__global__ __launch_bounds__(TPB)
void ud_zero_kernel(uint32_t* __restrict__ ov_max) {
  int n = blockIdx.x * TPB + threadIdx.x;
  if (n < NPB) ov_max[n] = 0u;   // IoU >= 0, so 0.0f bit pattern is identity
}

// ---- kernel 1: chunked pairwise IoU with async-LDS staging ----------------
__global__ __launch_bounds__(TPB)
void ud_iou_kernel(const float* __restrict__ people_boxes,   // [16,64,4]
                   const float* __restrict__ boxes,          // [50000,4]
                   const int*   __restrict__ image_index,    // [16]
                   uint32_t*    __restrict__ ov_max) {       // [1024]
  __shared__ float sbox[CHUNK * 4];                          // 16 KB

  const int tid   = threadIdx.x;
  const int n     = blockIdx.x * TPB + tid;                  // people box id
  const int base  = blockIdx.y * CHUNK;                      // box chunk base
  int valid = M_BOXES - base;
  if (valid > CHUNK) valid = CHUNK;

  // Stage this chunk into LDS: 4 x b128 per lane, tracked by ASYNCcnt.
#pragma unroll
  for (int k = 0; k < CHUNK / TPB; ++k) {
    int j  = tid + k * TPB;
    int gj = base + j;
    if (gj >= M_BOXES) gj = M_BOXES - 1;                     // clamp (dupes OK)
    async_copy_b128(boxes + 4 * gj, sbox + 4 * j);
  }

  // Overlap: transform this thread's people box while the DMA is in flight.
  const int img = n >> 6;                                    // n / 64
  const int lab = n & 63;
  const int idx = image_index[img];
  const int src = idx & (BATCH - 1);                         // concat-dup halves
  const float4 pb = ((const float4*)people_boxes)[src * MAX_LAB + lab];
  const bool nonzero = (pb.x + pb.y + pb.z + pb.w) != 0.0f;
  const float offx = nonzero ? IMG_W * (float)(img & 3) : 0.0f;   // (i/64)%4
  const float offy = nonzero ? IMG_H * (float)(img >> 2) : 0.0f;  // i/256
  const float cx = pb.x * IMG_W + offx;
  const float cy = pb.y * IMG_H + offy;
  const float bw = IMG_W * pb.z;
  const float bh = IMG_H * pb.w;
  const float px0 = cx - 0.5f * bw;
  const float py0 = cy - 0.5f * bh;
  const float px1 = px0 + bw;
  const float py1 = py0 + bh;
  const float area_p = (bw + 1.0f) * (bh + 1.0f);

  wait_async_zero();     // my wave's async copies landed in LDS
  __syncthreads();       // everyone's copies visible workgroup-wide

  // Division-free argmax: keep (num, den); compare a/b > c/d as a*d > c*b.
  float bn = 0.0f, bd = 1.0f;
  const float4* sb = (const float4*)sbox;
#pragma unroll 4
  for (int j = 0; j < valid; ++j) {
    const float4 b = sb[j];                                  // broadcast read
    const float ixm = fmaxf(px0, b.x);
    const float iym = fmaxf(py0, b.y);
    const float ixM = fminf(px1, b.z);
    const float iyM = fminf(py1, b.w);
    const float iw  = fmaxf(ixM - ixm + 1.0f, 0.0f);
    const float ih  = fmaxf(iyM - iym + 1.0f, 0.0f);
    const float inter  = iw * ih;
    const float area_b = (b.z - b.x + 1.0f) * (b.w - b.y + 1.0f);
    const float uni    = area_p + area_b - inter;
    if (inter * bd > bn * uni) { bn = inter; bd = uni; }
  }
  const float best = bn / bd;

  // uint-order == float-order for non-negative floats
  atomicMax(&ov_max[n], __float_as_uint(best));
}

// ---- kernel 2: finalize outputs -------------------------------------------
__global__ __launch_bounds__(TPB)
void ud_final_kernel(const uint32_t* __restrict__ ov_max,
                     const int*      __restrict__ image_index,
                     float*          __restrict__ out) {
  int n = blockIdx.x * TPB + threadIdx.x;
  if (n >= NPB) return;
  const float ov = __uint_as_float(ov_max[n]);
  out[2 * n]     = ov * 10.0f;            // overlaps / 0.1
  out[2 * n + 1] = (1.0f - ov) * 10.0f;   // (1 - overlaps) / 0.1
  out[2 * NPB + n] = (image_index[n >> 6] >= BATCH) ? 1.0f : 0.0f;
}

extern "C" void kernel_launch(void* const* d_in, const int* in_sizes, int n_in,
                              void* d_out, int out_size, void* d_ws, size_t ws_size,
                              hipStream_t stream) {
  const float* people_boxes = (const float*)d_in[0];   // [16*64*4]
  const float* boxes        = (const float*)d_in[1];   // [50000*4]
  const int*   image_index  = (const int*)d_in[2];     // [16]
  uint32_t*    ov_max       = (uint32_t*)d_ws;         // 1024 * 4B scratch
  float*       out          = (float*)d_out;           // 3072 floats

  ud_zero_kernel<<<dim3(NPB / TPB), dim3(TPB), 0, stream>>>(ov_max);

  dim3 grid(NPB / TPB, (M_BOXES + CHUNK - 1) / CHUNK); // 4 x 49 = 196 WGs
  ud_iou_kernel<<<grid, dim3(TPB), 0, stream>>>(people_boxes, boxes,
                                                image_index, ov_max);

  ud_final_kernel<<<dim3(NPB / TPB), dim3(TPB), 0, stream>>>(ov_max,
                                                             image_index, out);
}